// Neuron_28269474742650
// MI455X (gfx1250) — compile-verified
//
#include <hip/hip_runtime.h>
#include <math.h>

// ---------------------------------------------------------------------------
// Neuron stochastic-release reduction for MI455X (gfx1250, wave32).
//   total = sum_i [u_i < 0.9] * w_i * fatigue_i * x_i
//   out   = total > threshold ? tanh(total) : 0
// Bandwidth-bound: 256 MB streamed @ 23.3 TB/s ~= 11 us floor.
// Cross-lane reduction implemented with V_WMMA_F32_16X16X4_F32 (fp32-exact).
// ---------------------------------------------------------------------------

typedef __attribute__((ext_vector_type(2))) float v2f;
typedef __attribute__((ext_vector_type(4))) float v4f;
typedef __attribute__((ext_vector_type(8))) float v8f;

#define RELEASE_PROB 0.9f

// Reduce 32 per-lane fp32 partials to a single wave total using one
// f32 WMMA (A = partials in 16x4 layout with K=1/K=3 zeroed, B = ones).
// D[m][n] = acc(lane m) + acc(lane m+16) for every column n.
// Summing the 8 D VGPRs per lane gives rows 0-7 (lanes 0-15) and rows 8-15
// (lanes 16-31); one xor-16 shuffle completes the 32-lane sum.
// Requires EXEC == all ones (call from convergent code only).
__device__ __forceinline__ float wave_reduce_wmma_f32(float acc) {
  v2f a;
  a[0] = acc;   // lanes 0-15: A[m][0]; lanes 16-31: A[m][2]
  a[1] = 0.0f;  // lanes 0-15: A[m][1]; lanes 16-31: A[m][3]
  v2f b;
  b[0] = 1.0f;
  b[1] = 1.0f;  // B = all-ones 4x16 (layout irrelevant for constant matrix)
  v8f c = {};
  c = __builtin_amdgcn_wmma_f32_16x16x4_f32(
      /*neg_a=*/false, a, /*neg_b=*/false, b,
      /*c_mod=*/(short)0, c, /*reuse_a=*/false, /*reuse_b=*/false);
  float s = ((c[0] + c[1]) + (c[2] + c[3])) + ((c[4] + c[5]) + (c[6] + c[7]));
  s += __shfl_xor(s, 16, 32);  // combine rows 0-7 half with rows 8-15 half
  return s;                    // full wave total in every lane
}

// Block reduction: WMMA wave reduce, then LDS combine across (<=32) waves.
// Result valid on threadIdx.x == 0 only.
__device__ __forceinline__ float block_reduce(float acc, float* lds) {
  float wsum = wave_reduce_wmma_f32(acc);
  const int wave = threadIdx.x >> 5;
  const int lane = threadIdx.x & 31;
  if (lane == 0) lds[wave] = wsum;
  __syncthreads();
  float total = 0.0f;
  if (threadIdx.x == 0) {
    const int nw = blockDim.x >> 5;
    for (int i = 0; i < nw; ++i) total += lds[i];
  }
  return total;
}

__global__ void neuron_partial_kernel(const float* __restrict__ xp,
                                      const float* __restrict__ wp,
                                      const float* __restrict__ fp,
                                      const float* __restrict__ up,
                                      float* __restrict__ partials,
                                      int n4, int n) {
  __shared__ float lds[32];
  const v4f* __restrict__ x4 = (const v4f*)xp;
  const v4f* __restrict__ w4 = (const v4f*)wp;
  const v4f* __restrict__ f4 = (const v4f*)fp;
  const v4f* __restrict__ u4 = (const v4f*)up;

  float acc = 0.0f;
  const int stride = gridDim.x * blockDim.x;
  for (int i = blockIdx.x * blockDim.x + threadIdx.x; i < n4; i += stride) {
    // Non-temporal b128 streaming loads: data is single-use and exceeds L2.
    v4f xv = __builtin_nontemporal_load(&x4[i]);
    v4f wv = __builtin_nontemporal_load(&w4[i]);
    v4f fv = __builtin_nontemporal_load(&f4[i]);
    v4f uv = __builtin_nontemporal_load(&u4[i]);
#pragma unroll
    for (int k = 0; k < 4; ++k) {
      // mask * w * fatigue * x ; lowers to v_cndmask + v_fma (no EXEC change)
      float term = wv[k] * fv[k] * xv[k];
      acc += (uv[k] < RELEASE_PROB) ? term : 0.0f;
    }
  }

  // Scalar tail for N not divisible by 4 (reconverges before the reduce).
  if (blockIdx.x == 0 && threadIdx.x == 0) {
    for (int i = n4 * 4; i < n; ++i) {
      if (up[i] < RELEASE_PROB) acc += wp[i] * fp[i] * xp[i];
    }
  }

  float total = block_reduce(acc, lds);
  if (threadIdx.x == 0) partials[blockIdx.x] = total;  // every slot rewritten
}

__global__ void neuron_final_kernel(const float* __restrict__ partials,
                                    int np,
                                    const float* __restrict__ threshold,
                                    float* __restrict__ out) {
  __shared__ float lds[32];
  float acc = 0.0f;
  for (int i = threadIdx.x; i < np; i += blockDim.x) acc += partials[i];
  float total = block_reduce(acc, lds);
  if (threadIdx.x == 0) {
    const float t = threshold[0];
    out[0] = (total > t) ? tanhf(total) : 0.0f;
  }
}

extern "C" void kernel_launch(void* const* d_in, const int* in_sizes, int n_in,
                              void* d_out, int out_size, void* d_ws, size_t ws_size,
                              hipStream_t stream) {
  (void)n_in; (void)out_size; (void)ws_size;
  const float* x   = (const float*)d_in[0];
  const float* w   = (const float*)d_in[1];
  const float* fat = (const float*)d_in[2];
  const float* u   = (const float*)d_in[3];
  const float* thr = (const float*)d_in[4];
  float* out = (float*)d_out;
  float* partials = (float*)d_ws;  // needs <= 2048 floats (8 KB)

  const int n  = in_sizes[0];
  const int n4 = n >> 2;
  const int threads = 256;  // 8 wave32 per block
  int blocks = (n4 + threads - 1) / threads;
  if (blocks > 2048) blocks = 2048;
  if (blocks < 1) blocks = 1;

  neuron_partial_kernel<<<blocks, threads, 0, stream>>>(x, w, fat, u, partials, n4, n);
  neuron_final_kernel<<<1, 256, 0, stream>>>(partials, blocks, thr, out);
}